// TreeFFN_4466765988336
// MI455X (gfx1250) — compile-verified
//
#include <hip/hip_runtime.h>

typedef float v2f __attribute__((ext_vector_type(2)));
typedef float v8f __attribute__((ext_vector_type(8)));

#define DH 128

static __device__ __forceinline__ v8f wmma4(v2f a, v2f b, v8f c) {
  // V_WMMA_F32_16X16X4_F32 : D[16x16] = A[16x4] * B[4x16] + C, full fp32
  return __builtin_amdgcn_wmma_f32_16x16x4_f32(false, a, false, b, (short)0, c,
                                               false, false);
}

// ---------------------------------------------------------------------------
// One wave computes a 16-row x 128-col tile of  Out = f(X @ W^T [, Hres]).
// W is row-major [128 x K], so B[k][n] = W[n*K + k].
// MODE 0: Out = X@W^T          (initial projection h = node_feats @ W_s^T)
// MODE 1: Out = h + w*relu(X@W^T + h)   with w = sigmoid(T - step)
// ---------------------------------------------------------------------------
template <int MODE, int K>
__global__ __launch_bounds__(32) void gemm16_kernel(
    const float* __restrict__ X,     // [nrows, K]
    const float* __restrict__ W,     // [128, K]
    const float* Hres,               // [nrows, 128] (MODE==1; may alias Out)
    float* Out,                      // [nrows, 128]
    const float* __restrict__ Tptr,  // scalar T (MODE==1)
    int step, int nrows) {
  const int lane = threadIdx.x;        // 0..31
  const int lr = lane & 15;
  const int kh = (lane >> 4) << 1;     // lanes 0-15 -> K {0,1}; 16-31 -> {2,3}
  const int m0 = blockIdx.x * 16;

  v8f acc[8] = {};  // 8 N-tiles of 16 -> full 128 cols

  const int arow = m0 + lr;
  const bool aok = arow < nrows;
  const float* aptr = X + (size_t)(aok ? arow : 0) * K;

  for (int k = 0; k < K; k += 4) {
    v2f a;
    a.x = aok ? aptr[k + kh] : 0.f;
    a.y = aok ? aptr[k + kh + 1] : 0.f;
#pragma unroll
    for (int n = 0; n < 8; ++n) {
      const float* bp = W + (size_t)(n * 16 + lr) * K + (k + kh);
      v2f b = *(const v2f*)bp;
      acc[n] = wmma4(a, b, acc[n]);
    }
  }

  float wstep = 0.f;
  if (MODE == 1) {
    float T = Tptr[0];
    wstep = 1.f / (1.f + __expf((float)step - T));
  }

  const int rbase = (lane >> 4) << 3;  // C/D: VGPR r -> row r + 8*(lane>=16)
#pragma unroll
  for (int n = 0; n < 8; ++n) {
#pragma unroll
    for (int r = 0; r < 8; ++r) {
      int row = m0 + rbase + r;
      if (row < nrows) {
        size_t idx = (size_t)row * DH + (n * 16 + lr);
        float v = acc[n][r];
        if (MODE == 1) {
          float hv = Hres[idx];
          float t = v + hv;
          Out[idx] = hv + wstep * (t > 0.f ? t : 0.f);
        } else {
          Out[idx] = v;
        }
      }
    }
  }
}

// ---------------------------------------------------------------------------
// One wave handles 16 edges:
//   ec[16,256] = [h[p] | h[c]]  (gathered to LDS, padded rows)
//   alpha[16]  = sigmoid(ec @ gw + gb)
//   msg[16,128]= (ec @ W_edge^T) * alpha      (512x V_WMMA_F32_16X16X4_F32)
//   agg[p] += msg ; agg[c] += msg             (hw fp32 atomics, L2-resident)
// ---------------------------------------------------------------------------
#define EC_PAD 260  // 260 % 64 banks == 4 -> 16 rows land on distinct banks

__global__ __launch_bounds__(32) void edge_kernel(
    const float* __restrict__ H,    // [N,128]
    const int* __restrict__ Pidx,   // [E]
    const int* __restrict__ Cidx,   // [E]
    const float* __restrict__ We,   // [128,256] row-major
    const float* __restrict__ Gw,   // [256]
    const float* __restrict__ Gb,   // [1]
    float* __restrict__ Agg,        // [N,128]
    int E) {
  __shared__ float ec[16][EC_PAD];
  __shared__ float alpha_s[16];
  __shared__ int pc_s[32];  // p[0..15], c[0..15]

  const int lane = threadIdx.x;
  const int e0 = blockIdx.x * 16;

  if (lane < 16) {
    int e = e0 + lane;
    int ee = (e < E) ? e : e0;  // clamp; masked to zero via alpha below
    pc_s[lane] = Pidx[ee];
    pc_s[16 + lane] = Cidx[ee];
  }
  __syncthreads();  // single-wave WG: barrier is a NOP, keeps LDS ordering

  // Gather 16x(128+128) feature rows, coalesced float4 per lane.
  for (int i = 0; i < 16; ++i) {
    const float4* hp = (const float4*)(H + (size_t)pc_s[i] * DH);
    const float4* hc = (const float4*)(H + (size_t)pc_s[16 + i] * DH);
    float4 vp = hp[lane];
    float4 vc = hc[lane];
    *(float4*)&ec[i][lane * 4] = vp;
    *(float4*)&ec[i][128 + lane * 4] = vc;
  }
  __syncthreads();

  // Gate: alpha[i] = sigmoid(dot(ec[i,:], gw) + gb), wave-wide reduction.
  float gw[8];
#pragma unroll
  for (int j = 0; j < 8; ++j) gw[j] = Gw[lane + 32 * j];
  float gb = Gb[0];
  for (int i = 0; i < 16; ++i) {
    float part = 0.f;
#pragma unroll
    for (int j = 0; j < 8; ++j) part += ec[i][lane + 32 * j] * gw[j];
#pragma unroll
    for (int off = 16; off > 0; off >>= 1) part += __shfl_xor(part, off, 32);
    if (lane == 0) {
      bool valid = (e0 + i) < E;
      alpha_s[i] = valid ? (1.f / (1.f + __expf(-(part + gb)))) : 0.f;
    }
  }
  __syncthreads();

  // msg = ec @ We^T  via fp32 WMMA, K=256, N=128 in 8 tiles.
  const int lr = lane & 15;
  const int kh = (lane >> 4) << 1;
  v8f acc[8] = {};
  for (int k = 0; k < 256; k += 4) {
    v2f a = *(const v2f*)&ec[lr][k + kh];
#pragma unroll
    for (int n = 0; n < 8; ++n) {
      const float* bp = We + (size_t)(n * 16 + lr) * 256 + (k + kh);
      v2f b = *(const v2f*)bp;
      acc[n] = wmma4(a, b, acc[n]);
    }
  }

  // Scale by gate and scatter-add into both endpoints.
  const int rbase = (lane >> 4) << 3;
  float al[8];
  int pn[8], cn[8];
#pragma unroll
  for (int r = 0; r < 8; ++r) {
    al[r] = alpha_s[rbase + r];
    pn[r] = pc_s[rbase + r];
    cn[r] = pc_s[16 + rbase + r];
  }
#pragma unroll
  for (int n = 0; n < 8; ++n) {
    const int col = n * 16 + lr;
#pragma unroll
    for (int r = 0; r < 8; ++r) {
      float v = acc[n][r] * al[r];  // alpha==0 for clamped tail rows
      unsafeAtomicAdd(&Agg[(size_t)pn[r] * DH + col], v);
      unsafeAtomicAdd(&Agg[(size_t)cn[r] * DH + col], v);
    }
  }
}

__global__ void clear_kernel(float4* __restrict__ p, int n4) {
  int i = blockIdx.x * blockDim.x + threadIdx.x;
  if (i < n4) p[i] = make_float4(0.f, 0.f, 0.f, 0.f);
}

extern "C" void kernel_launch(void* const* d_in, const int* in_sizes, int n_in,
                              void* d_out, int out_size, void* d_ws,
                              size_t ws_size, hipStream_t stream) {
  const float* node_feats = (const float*)d_in[0];
  const int* edge_index = (const int*)d_in[1];
  const float* W_s = (const float*)d_in[2];
  const float* W_pc = (const float*)d_in[3];
  const float* W_edge = (const float*)d_in[4];
  const float* W_gate_w = (const float*)d_in[5];
  const float* W_gate_b = (const float*)d_in[6];
  const float* Tptr = (const float*)d_in[7];

  const int N = in_sizes[0] / DH;  // 100000
  const int E = in_sizes[1] / 2;   // 500000
  const int* p_idx = edge_index;
  const int* c_idx = edge_index + E;

  float* h = (float*)d_out;   // h lives in d_out; fully rewritten by stage 0
  float* agg = (float*)d_ws;  // N*128 f32 scratch (51.2 MB, L2-resident)

  const int node_tiles = (N + 15) / 16;
  const int edge_tiles = (E + 15) / 16;
  const int n4 = N * DH / 4;

  // h = node_feats @ W_s^T
  gemm16_kernel<0, DH><<<node_tiles, 32, 0, stream>>>(node_feats, W_s, nullptr,
                                                      h, nullptr, 0, N);

  for (int step = 0; step < 3; ++step) {
    clear_kernel<<<(n4 + 255) / 256, 256, 0, stream>>>((float4*)agg, n4);
    edge_kernel<<<edge_tiles, 32, 0, stream>>>(h, p_idx, c_idx, W_edge,
                                               W_gate_w, W_gate_b, agg, E);
    // h = h + sigmoid(T-step) * relu(agg @ W_pc^T + h)   (in place)
    gemm16_kernel<1, DH><<<node_tiles, 32, 0, stream>>>(agg, W_pc, h, h, Tptr,
                                                        step, N);
  }
}